// VectorQuantizer_35081292873933
// MI455X (gfx1250) — compile-verified
//
#include <hip/hip_runtime.h>

typedef __attribute__((ext_vector_type(2))) float v2f;
typedef __attribute__((ext_vector_type(8))) float v8f;

#define T_TOKENS 32768      // 16 * 2048
#define DIM 256
#define K_CODES 8192
#define KTILES (K_CODES / 16)
#define NW 8                // waves per block in argmin kernel
#define BROW 260            // padded LDS row stride in floats (bank-conflict-free)
#define BTILE (16 * BROW)   // floats per LDS B tile

// low 32 bits of a flat shared-memory address == LDS byte offset
__device__ __forceinline__ unsigned lds_off(const void* p) {
    return (unsigned)(size_t)p;
}

// one 16-byte async global->LDS copy (gfx1250, tracked by ASYNCcnt)
__device__ __forceinline__ void async_ld16(unsigned ldsoff, unsigned goff,
                                           unsigned long long gbase) {
    asm volatile("global_load_async_to_lds_b128 %0, %1, %2"
                 :: "v"(ldsoff), "v"(goff), "s"(gbase)
                 : "memory");
}

__device__ __forceinline__ void wait_async0() {
    asm volatile("s_wait_asynccnt 0x0" ::: "memory");
}

// ---------------------------------------------------------------------------
// w2[k] = sum_d codebook[k][d]^2.  One wave per row, lane-strided.
// ---------------------------------------------------------------------------
__global__ __launch_bounds__(256) void vq_w2_kernel(const float* __restrict__ cb,
                                                    float* __restrict__ w2) {
    int gwave = (blockIdx.x * blockDim.x + threadIdx.x) >> 5;
    int lane  = threadIdx.x & 31;
    if (gwave >= K_CODES) return;
    const float* row = cb + (size_t)gwave * DIM;
    float s = 0.f;
#pragma unroll
    for (int i = 0; i < DIM / 32; ++i) {
        float v = row[lane + 32 * i];
        s += v * v;
    }
#pragma unroll
    for (int m = 16; m >= 1; m >>= 1) s += __shfl_xor(s, m, 32);
    if (lane == 0) w2[gwave] = s;
}

// ---------------------------------------------------------------------------
__global__ void vq_zero_used_kernel(int* __restrict__ used) {
    int i = blockIdx.x * blockDim.x + threadIdx.x;
    if (i < K_CODES) used[i] = 0;
}

// ---------------------------------------------------------------------------
// Argmin over codes via fp32 WMMA with async-LDS double-buffered codebook.
// Block: 8 waves, each owning one 16-token tile (A in 128 VGPRs).
// All waves sweep the same code tile; the 16x256 fp32 codebook tile is
// async-copied to padded LDS and shared by every wave.
//
// A frag (16x4 f32): lanes 0-15 -> M=lane, K=d0+{0,1}; lanes 16-31 -> K=d0+{2,3}
// B frag (4x16 f32): lanes 0-15 -> N=lane, K=d0+{0,1}; lanes 16-31 -> K=d0+{2,3}
// C/D (16x16 f32):   VGPR r: lanes 0-15 -> (M=r, N=lane); lanes 16-31 -> (M=r+8)
// ---------------------------------------------------------------------------
__global__ __launch_bounds__(NW * 32) void vq_argmin_kernel(
    const float* __restrict__ z, const float* __restrict__ cb,
    const float* __restrict__ w2, int* __restrict__ out_idx) {

    __shared__ __align__(16) float bbuf[2][BTILE];

    const int tid  = threadIdx.x;
    const int lane = tid & 31;
    const int wave = tid >> 5;
    const int half = lane >> 4;           // 0: lanes 0-15, 1: lanes 16-31
    const int l16  = lane & 15;
    const int tile = blockIdx.x * NW + wave;   // this wave's token tile

    // ---- A tile: 16 tokens x 256 dims resident in registers (64 v2f frags)
    const v2f* zrow = (const v2f*)(z + (size_t)(tile * 16 + l16) * DIM + half * 2);
    v2f a[64];
#pragma unroll
    for (int f = 0; f < 64; ++f) a[f] = zrow[f * 2];   // stride 16B

    // ---- async-copy addressing: 1024 x 16B chunks per tile, 4 per thread
    const int chunk0 = tid * 4;                 // 4 consecutive chunks, same row
    const int brow   = chunk0 >> 6;             // 64 chunks per 256-float row
    const int bcol   = (chunk0 & 63) * 4;       // float column of first chunk
    const unsigned goff = (unsigned)((brow * DIM + bcol) * 4);
    const unsigned lds0 = lds_off(&bbuf[0][brow * BROW + bcol]);
    const unsigned lds1 = lds_off(&bbuf[1][brow * BROW + bcol]);
    const unsigned long long cbase = (unsigned long long)(size_t)cb;

    float minv[8];
    int   mini[8];
#pragma unroll
    for (int r = 0; r < 8; ++r) { minv[r] = 3.4e38f; mini[r] = 0x7fffffff; }

    // prologue: prefetch tile 0 into buffer 0
    {
        unsigned long long gb = cbase;          // ct = 0
#pragma unroll
        for (int j = 0; j < 4; ++j) async_ld16(lds0 + j * 16u, goff + j * 16u, gb);
    }

    for (int ct = 0; ct < KTILES; ++ct) {
        wait_async0();        // our chunks of tile ct have landed in LDS
        __syncthreads();      // all chunks landed; all waves done with tile ct-1

        if (ct + 1 < KTILES) {   // prefetch next tile into the other buffer
            unsigned long long gb = cbase + (unsigned long long)(ct + 1) * (16 * DIM * 4);
            unsigned lbase = ((ct + 1) & 1) ? lds1 : lds0;
#pragma unroll
            for (int j = 0; j < 4; ++j) async_ld16(lbase + j * 16u, goff + j * 16u, gb);
        }

        const v2f* bfrag = (const v2f*)(&bbuf[ct & 1][l16 * BROW + half * 2]);
        v8f acc = {};
#pragma unroll
        for (int f = 0; f < 64; ++f) {
            acc = __builtin_amdgcn_wmma_f32_16x16x4_f32(
                false, a[f], false, bfrag[f * 2], (short)0, acc, false, false);
        }

        const int   c   = ct * 16 + l16;        // this lane's code column
        const float w2c = w2[c];
#pragma unroll
        for (int r = 0; r < 8; ++r) {
            float dist = __builtin_fmaf(-2.0f, acc[r], w2c);   // w2 - 2*dot
            bool lt = dist < minv[r];
            minv[r] = lt ? dist : minv[r];
            mini[r] = lt ? c    : mini[r];
        }
    }

    // Reduce across the 16 code columns held inside each half-wave.
#pragma unroll
    for (int r = 0; r < 8; ++r) {
#pragma unroll
        for (int m = 8; m >= 1; m >>= 1) {
            float ov = __shfl_xor(minv[r], m, 32);
            int   oi = __shfl_xor(mini[r], m, 32);
            bool take = (ov < minv[r]) || (ov == minv[r] && oi < mini[r]);
            minv[r] = take ? ov : minv[r];
            mini[r] = take ? oi : mini[r];
        }
    }

    // lane 0 holds tokens m=0..7, lane 16 holds tokens m=8..15
    if (l16 == 0) {
#pragma unroll
        for (int r = 0; r < 8; ++r)
            out_idx[tile * 16 + half * 8 + r] = mini[r];
    }
}

// ---------------------------------------------------------------------------
// z_q = codebook[idx]; per-token partial sum of (z - z_q)^2; usage scatter.
// One block (256 threads) per token. No atomics -> deterministic.
// ---------------------------------------------------------------------------
__global__ __launch_bounds__(256) void vq_gather_loss_kernel(
    const float* __restrict__ z, const float* __restrict__ cb,
    const int* __restrict__ idx, float* __restrict__ zq_out,
    float* __restrict__ partials, int* __restrict__ used) {
    const int t = blockIdx.x;
    const int d = threadIdx.x;
    const int k = idx[t];
    float q = cb[(size_t)k * DIM + d];
    float e = z[(size_t)t * DIM + d];
    zq_out[(size_t)t * DIM + d] = q;
    float diff = e - q;

    __shared__ float red[256];
    red[d] = diff * diff;
    __syncthreads();
#pragma unroll
    for (int s = 128; s > 0; s >>= 1) {
        if (d < s) red[d] += red[d + s];
        __syncthreads();
    }
    if (d == 0) {
        partials[t] = red[0];
        used[k] = 1;                    // same-value races are benign
    }
}

// ---------------------------------------------------------------------------
// Final fixed-order reduction: loss = 1.25 * mean(diff^2); usage = nnz/K.
// ---------------------------------------------------------------------------
__global__ __launch_bounds__(1024) void vq_finalize_kernel(
    const float* __restrict__ partials, const int* __restrict__ used,
    float* __restrict__ out_tail) {
    __shared__ float redf[1024];
    __shared__ int   redi[1024];
    const int tid = threadIdx.x;

    float s = 0.f;
    for (int i = tid; i < T_TOKENS; i += 1024) s += partials[i];
    int u = 0;
    for (int i = tid; i < K_CODES; i += 1024) u += used[i];
    redf[tid] = s;
    redi[tid] = u;
    __syncthreads();
#pragma unroll
    for (int st = 512; st > 0; st >>= 1) {
        if (tid < st) {
            redf[tid] += redf[tid + st];
            redi[tid] += redi[tid + st];
        }
        __syncthreads();
    }
    if (tid == 0) {
        out_tail[0] = redf[0] * (1.25f / (float)(T_TOKENS * DIM));
        out_tail[1] = (float)redi[0] / (float)K_CODES;
    }
}

// ---------------------------------------------------------------------------
extern "C" void kernel_launch(void* const* d_in, const int* in_sizes, int n_in,
                              void* d_out, int out_size, void* d_ws, size_t ws_size,
                              hipStream_t stream) {
    const float* z  = (const float*)d_in[0];   // [16,2048,256] fp32
    const float* cb = (const float*)d_in[1];   // [8192,256] fp32
    float* out = (float*)d_out;                // [8388608 zq | loss | usage]

    char* ws = (char*)d_ws;
    int*   idx      = (int*)(ws);                      // 128 KB
    float* w2       = (float*)(ws + 128 * 1024);       //  32 KB
    float* partials = (float*)(ws + 160 * 1024);       // 128 KB
    int*   used     = (int*)(ws + 288 * 1024);         //  32 KB

    vq_zero_used_kernel<<<K_CODES / 256, 256, 0, stream>>>(used);
    vq_w2_kernel<<<K_CODES / 8, 256, 0, stream>>>(cb, w2);
    vq_argmin_kernel<<<T_TOKENS / (16 * NW), NW * 32, 0, stream>>>(z, cb, w2, idx);
    vq_gather_loss_kernel<<<T_TOKENS, 256, 0, stream>>>(z, cb, idx, out, partials, used);
    vq_finalize_kernel<<<1, 1024, 0, stream>>>(partials, used, out + (size_t)T_TOKENS * DIM);
}